// AttnDecoderRNN_6081673691393
// MI455X (gfx1250) — compile-verified
//
#include <hip/hip_runtime.h>
#include <hip/hip_bf16.h>
#include <math.h>

#define H 2048
#define V 50257
#define L 512

typedef __attribute__((ext_vector_type(2))) float v2f;
typedef __attribute__((ext_vector_type(8))) float v8f;

// ---- workspace layout (float offsets), total ~320 KB ----
#define WS_XCAT1   0        // 4096  : [embedded | h0]
#define WS_ALOGITS 4096     // 512   : attention logits
#define WS_AWEIGHT 4608     // 512   : attention weights
#define WS_AAPPL   5120     // 2048  : attn_applied
#define WS_XCAT2   8192     // 4096  : [embedded | attn_applied]
#define WS_X       12288    // 2048  : relu(combined)
#define WS_GI      14336    // 6144  : W_ih @ x + b_ih
#define WS_GH      20480    // 6144  : W_hh @ h0 + b_hh
#define WS_H1      26624    // 2048  : new hidden
#define WS_LOGITS  28672    // 50257 : vocab logits
#define WS_RED     79936    // 2     : {max, log(sum exp)}

// ============================================================
// y[r] = sum_k W[r,k] * x[k] + b[r]   (optionally ReLU)
// One block = 16 output rows; 4 waves split K (K % 64 == 0).
// V_WMMA_F32_16X16X4_F32:
//   A (16x4, documented layout): lane<16 -> row=lane,    K={k,k+1}
//                                lane>=16 -> row=lane-16, K={k+2,k+3}
//   B (4x16): x broadcast into all 16 columns, same K split,
//             so every D column equals the matvec partial.
// D layout: vgpr p -> M=p (lanes 0-15) / M=8+p (lanes 16-31).
// Weights are streamed once -> non-temporal loads (protect L2
// for x, which is re-read by every block).
// ============================================================
__global__ __launch_bounds__(128)
void matvec_wmma_f32(const float* __restrict__ Wm, const float* __restrict__ x,
                     const float* __restrict__ bias, float* __restrict__ y,
                     int R, int K, int relu_flag) {
    __shared__ float red[4][16];
    const int tid  = threadIdx.x;
    // scalarize the wave id -> uniform loop bounds, EXEC stays all-ones
    const int wave = __builtin_amdgcn_readfirstlane(tid >> 5);
    const int lane = tid & 31;
    const int r0   = blockIdx.x << 4;
    int row = r0 + (lane & 15);
    if (row >= R) row = R - 1;              // clamp loads; masked at store
    const int khalf = (lane >> 4) << 1;     // 0 or 2
    const int Kq    = K >> 2;               // per-wave K span (mult of 16)
    const int kbeg  = wave * Kq;

    const float* wp = Wm + (size_t)row * K + kbeg + khalf;
    const float* xp = x + kbeg + khalf;

    v8f c0 = {0.f,0.f,0.f,0.f,0.f,0.f,0.f,0.f};
    v8f c1 = {0.f,0.f,0.f,0.f,0.f,0.f,0.f,0.f};

    const int niter = Kq >> 4;              // 16 K-values / iteration
    for (int it = 0; it < niter; ++it) {
        v2f a0 = __builtin_nontemporal_load(reinterpret_cast<const v2f*>(wp));
        v2f a1 = __builtin_nontemporal_load(reinterpret_cast<const v2f*>(wp + 4));
        v2f a2 = __builtin_nontemporal_load(reinterpret_cast<const v2f*>(wp + 8));
        v2f a3 = __builtin_nontemporal_load(reinterpret_cast<const v2f*>(wp + 12));
        v2f b0 = *reinterpret_cast<const v2f*>(xp);
        v2f b1 = *reinterpret_cast<const v2f*>(xp + 4);
        v2f b2 = *reinterpret_cast<const v2f*>(xp + 8);
        v2f b3 = *reinterpret_cast<const v2f*>(xp + 12);
        c0 = __builtin_amdgcn_wmma_f32_16x16x4_f32(false, a0, false, b0,
                                                   (short)0, c0, false, false);
        c1 = __builtin_amdgcn_wmma_f32_16x16x4_f32(false, a1, false, b1,
                                                   (short)0, c1, false, false);
        c0 = __builtin_amdgcn_wmma_f32_16x16x4_f32(false, a2, false, b2,
                                                   (short)0, c0, false, false);
        c1 = __builtin_amdgcn_wmma_f32_16x16x4_f32(false, a3, false, b3,
                                                   (short)0, c1, false, false);
        wp += 16;
        xp += 16;
    }

    // lanes 0 and 16 of each wave hold M=0..7 / M=8..15 in the 8 acc VGPRs
    if ((tid & 15) == 0) {
        const int mh = (tid >> 4) & 1;
        #pragma unroll
        for (int p = 0; p < 8; ++p) red[wave][mh * 8 + p] = c0[p] + c1[p];
    }
    __syncthreads();
    if (tid < 16) {
        const int r = r0 + tid;
        if (r < R) {
            float v = red[0][tid] + red[1][tid] + red[2][tid] + red[3][tid] + bias[r];
            if (relu_flag) v = v > 0.f ? v : 0.f;
            y[r] = v;
        }
    }
}

// ---- small fused helpers ----------------------------------
__global__ void pack_xcat1(const int* __restrict__ tok, const float* __restrict__ emb,
                           const float* __restrict__ h0, float* __restrict__ ws) {
    const int i = blockIdx.x * blockDim.x + threadIdx.x;      // 0..4095
    const int t = tok[0];
    ws[WS_XCAT1 + i] = (i < H) ? emb[(size_t)t * H + i] : h0[i - H];
}

__global__ void pack_xcat2(float* __restrict__ ws) {
    const int i = blockIdx.x * blockDim.x + threadIdx.x;      // 0..4095
    ws[WS_XCAT2 + i] = (i < H) ? ws[WS_XCAT1 + i] : ws[WS_AAPPL + (i - H)];
}

__global__ __launch_bounds__(512)
void softmax512(const float* __restrict__ logits, float* __restrict__ wts,
                float* __restrict__ out_tail) {
    __shared__ float s[512];
    const int t = threadIdx.x;
    const float v = logits[t];
    s[t] = v; __syncthreads();
    for (int off = 256; off > 0; off >>= 1) {
        if (t < off) s[t] = fmaxf(s[t], s[t + off]);
        __syncthreads();
    }
    const float mx = s[0]; __syncthreads();
    const float e = expf(v - mx);
    s[t] = e; __syncthreads();
    for (int off = 256; off > 0; off >>= 1) {
        if (t < off) s[t] += s[t + off];
        __syncthreads();
    }
    const float w = e / s[0];
    wts[t] = w;
    out_tail[t] = w;
}

// attn_applied[h] = sum_l w[l] * E[l,h]  (column-coalesced across threads)
__global__ void attn_apply(const float* __restrict__ wts, const float* __restrict__ E,
                           float* __restrict__ y) {
    const int h = blockIdx.x * blockDim.x + threadIdx.x;
    float acc = 0.f;
    for (int l = 0; l < L; ++l) acc = fmaf(wts[l], E[(size_t)l * H + h], acc);
    y[h] = acc;
}

__global__ void gru_combine(const float* __restrict__ gi, const float* __restrict__ gh,
                            const float* __restrict__ h0, float* __restrict__ h1ws,
                            float* __restrict__ h1out) {
    const int h = blockIdx.x * blockDim.x + threadIdx.x;
    const float r = 1.f / (1.f + expf(-(gi[h]         + gh[h])));
    const float z = 1.f / (1.f + expf(-(gi[H + h]     + gh[H + h])));
    const float n = tanhf(gi[2 * H + h] + r * gh[2 * H + h]);
    const float v = (1.f - z) * n + z * h0[h];
    h1ws[h] = v;
    h1out[h] = v;
}

__global__ __launch_bounds__(1024)
void lse_reduce(const float* __restrict__ logits, float* __restrict__ red) {
    __shared__ float s[1024];
    const int t = threadIdx.x;
    float mx = -INFINITY;
    for (int i = t; i < V; i += 1024) mx = fmaxf(mx, logits[i]);
    s[t] = mx; __syncthreads();
    for (int off = 512; off > 0; off >>= 1) {
        if (t < off) s[t] = fmaxf(s[t], s[t + off]);
        __syncthreads();
    }
    mx = s[0]; __syncthreads();
    float sum = 0.f;
    for (int i = t; i < V; i += 1024) sum += expf(logits[i] - mx);
    s[t] = sum; __syncthreads();
    for (int off = 512; off > 0; off >>= 1) {
        if (t < off) s[t] += s[t + off];
        __syncthreads();
    }
    if (t == 0) { red[0] = mx; red[1] = logf(s[0]); }
}

__global__ void write_logprobs(const float* __restrict__ logits,
                               const float* __restrict__ red, float* __restrict__ out) {
    const int i = blockIdx.x * blockDim.x + threadIdx.x;
    if (i < V) out[i] = logits[i] - red[0] - red[1];
}

// ============================================================
extern "C" void kernel_launch(void* const* d_in, const int* in_sizes, int n_in,
                              void* d_out, int out_size, void* d_ws, size_t ws_size,
                              hipStream_t stream) {
    const int*   tok    = (const int*)  d_in[0];
    const float* hidden = (const float*)d_in[1];   // [1,1,H]
    const float* enc    = (const float*)d_in[2];   // [L,H]
    const float* emb    = (const float*)d_in[3];   // [V,H]
    const float* attn_W = (const float*)d_in[4];   // [L,2H]
    const float* attn_b = (const float*)d_in[5];
    const float* comb_W = (const float*)d_in[6];   // [H,2H]
    const float* comb_b = (const float*)d_in[7];
    const float* W_ih   = (const float*)d_in[8];   // [3H,H]
    const float* W_hh   = (const float*)d_in[9];   // [3H,H]
    const float* b_ih   = (const float*)d_in[10];
    const float* b_hh   = (const float*)d_in[11];
    const float* out_W  = (const float*)d_in[12];  // [V,H]
    const float* out_b  = (const float*)d_in[13];
    float* out = (float*)d_out;                    // [V | H | L]
    float* ws  = (float*)d_ws;

    // 1) xcat1 = [emb[tok], h0]
    pack_xcat1<<<16, 256, 0, stream>>>(tok, emb, hidden, ws);
    // 2) attn_logits = attn_W @ xcat1 + attn_b         (512 x 4096)
    matvec_wmma_f32<<<L / 16, 128, 0, stream>>>(attn_W, ws + WS_XCAT1, attn_b,
                                                ws + WS_ALOGITS, L, 2 * H, 0);
    // 3) softmax -> attn_weights (also final output tail)
    softmax512<<<1, 512, 0, stream>>>(ws + WS_ALOGITS, ws + WS_AWEIGHT, out + V + H);
    // 4) attn_applied = attn_weights @ encoder_outputs
    attn_apply<<<H / 256, 256, 0, stream>>>(ws + WS_AWEIGHT, enc, ws + WS_AAPPL);
    // 5) xcat2 = [embedded, attn_applied]
    pack_xcat2<<<16, 256, 0, stream>>>(ws);
    // 6) x = relu(comb_W @ xcat2 + comb_b)             (2048 x 4096)
    matvec_wmma_f32<<<H / 16, 128, 0, stream>>>(comb_W, ws + WS_XCAT2, comb_b,
                                                ws + WS_X, H, 2 * H, 1);
    // 7) GRU gate matvecs                              (6144 x 2048 each)
    matvec_wmma_f32<<<(3 * H) / 16, 128, 0, stream>>>(W_ih, ws + WS_X, b_ih,
                                                      ws + WS_GI, 3 * H, H, 0);
    matvec_wmma_f32<<<(3 * H) / 16, 128, 0, stream>>>(W_hh, hidden, b_hh,
                                                      ws + WS_GH, 3 * H, H, 0);
    // 8) h1 = GRU pointwise combine  -> output slot [V : V+H)
    gru_combine<<<H / 256, 256, 0, stream>>>(ws + WS_GI, ws + WS_GH, hidden,
                                             ws + WS_H1, out + V);
    // 9) vocab logits = out_W @ h1 + out_b             (50257 x 2048, 412 MB)
    matvec_wmma_f32<<<(V + 15) / 16, 128, 0, stream>>>(out_W, ws + WS_H1, out_b,
                                                       ws + WS_LOGITS, V, H, 0);
    // 10) log_softmax
    lse_reduce<<<1, 1024, 0, stream>>>(ws + WS_LOGITS, ws + WS_RED);
    write_logprobs<<<(V + 255) / 256, 256, 0, stream>>>(ws + WS_LOGITS, ws + WS_RED, out);
}